// SoftTopKBottomK_9242769621105
// MI455X (gfx1250) — compile-verified
//
#include <hip/hip_runtime.h>
#include <cstdint>

// SOFT top-k/bottom-k via scaling-domain Sinkhorn (m = 3 anchors).
// One block per row; k_ij kernel values live in registers for all 200 iters.

constexpr int   kN     = 4096;         // scores per row
constexpr int   kTPB   = 256;          // 8 wave32 waves
constexpr int   kWaves = kTPB / 32;
constexpr int   kEPT   = kN / kTPB;    // 16 elements per thread
constexpr int   kIters = 200;
constexpr float kInvEps = 10.0f;       // 1 / epsilon
// nu * n for anchors {0, 0.5, 1}, K_TOP = 512, N = 4096
constexpr float kNu0N = 512.0f;
constexpr float kNu1N = 3072.0f;
constexpr float kNu2N = 512.0f;

__global__ __launch_bounds__(kTPB) void soft_topk_sinkhorn(
    const float* __restrict__ scores, float* __restrict__ out) {
  __shared__ float xs[kN];
  __shared__ float red[2][kWaves][4];   // double-buffered cross-wave scratch

  const int row  = blockIdx.x;
  const int tid  = threadIdx.x;
  const int lane = tid & 31;
  const int wave = tid >> 5;
  const float* rowp = scores + (size_t)row * kN;

  // ---- Stage row (16 KB) HBM -> LDS with CDNA5 async DMA (ASYNCcnt) ----
  {
    auto* xs3 = (__attribute__((address_space(3))) float*)xs;
    #pragma unroll
    for (int c = 0; c < 4; ++c) {                       // 4 x b128 per thread
      unsigned voff = (unsigned)(tid * (kEPT * 4)) + (unsigned)(c * 16);
      unsigned vdst = (unsigned)(unsigned long)xs3 + voff;
      asm volatile("global_load_async_to_lds_b128 %0, %1, %2 offset:0"
                   :: "v"(vdst), "v"(voff), "s"(rowp)
                   : "memory");
    }
    asm volatile("s_wait_asynccnt 0x0" ::: "memory");
  }
  __syncthreads();

  // ---- Read my 16 contiguous elements; local min/max ----
  float x[kEPT];
  float lmin = 3.0e38f, lmax = -3.0e38f;
  {
    const float4* xv = (const float4*)(xs + tid * kEPT);
    #pragma unroll
    for (int c = 0; c < 4; ++c) {
      float4 t = xv[c];
      x[4 * c + 0] = t.x; x[4 * c + 1] = t.y;
      x[4 * c + 2] = t.z; x[4 * c + 3] = t.w;
    }
    #pragma unroll
    for (int e = 0; e < kEPT; ++e) {
      lmin = fminf(lmin, x[e]);
      lmax = fmaxf(lmax, x[e]);
    }
  }
  // wave32 reduce, then 8-wave LDS combine
  #pragma unroll
  for (int m = 16; m >= 1; m >>= 1) {
    lmin = fminf(lmin, __shfl_xor(lmin, m, 32));
    lmax = fmaxf(lmax, __shfl_xor(lmax, m, 32));
  }
  if (lane == 0) { red[0][wave][0] = lmin; red[0][wave][1] = lmax; }
  __syncthreads();
  float smin = red[0][0][0], smax = red[0][0][1];
  #pragma unroll
  for (int w = 1; w < kWaves; ++w) {
    smin = fminf(smin, red[0][w][0]);
    smax = fmaxf(smax, red[0][w][1]);
  }
  __syncthreads();  // red[0] is reused by iteration 0 below

  // ---- Precompute Gibbs kernel k_ij = exp(-(x-a_j)^2/eps), once ----
  const float scale = 1.0f / (smax - smin + 1e-12f);
  float k0[kEPT], k1[kEPT], k2[kEPT];
  #pragma unroll
  for (int e = 0; e < kEPT; ++e) {
    float xn = (x[e] - smin) * scale;
    float d0 = xn;          k0[e] = __expf(-kInvEps * d0 * d0);
    float d1 = xn - 0.5f;   k1[e] = __expf(-kInvEps * d1 * d1);
    float d2 = xn - 1.0f;   k2[e] = __expf(-kInvEps * d2 * d2);
  }

  // ---- 200 Sinkhorn iterations, no exp/log in the loop ----
  float w0 = 1.0f, w1 = 1.0f, w2 = 1.0f;   // exp(v_j)
  float p0 = 1.0f, p1 = 1.0f, p2 = 1.0f;   // previous w (for final denom)
  #pragma unroll 1
  for (int it = 0; it < kIters; ++it) {
    p0 = w0; p1 = w1; p2 = w2;
    float s0 = 0.0f, s1 = 0.0f, s2 = 0.0f;
    #pragma unroll
    for (int e = 0; e < kEPT; ++e) {
      float d = fmaf(k0[e], w0, fmaf(k1[e], w1, k2[e] * w2));
      float r = __builtin_amdgcn_rcpf(d);         // 1/denom_i
      s0 = fmaf(k0[e], r, s0);
      s1 = fmaf(k1[e], r, s1);
      s2 = fmaf(k2[e], r, s2);
    }
    #pragma unroll
    for (int m = 16; m >= 1; m >>= 1) {
      s0 += __shfl_xor(s0, m, 32);
      s1 += __shfl_xor(s1, m, 32);
      s2 += __shfl_xor(s2, m, 32);
    }
    const int slot = it & 1;
    if (lane == 0) {
      red[slot][wave][0] = s0; red[slot][wave][1] = s1; red[slot][wave][2] = s2;
    }
    __syncthreads();
    float t0 = 0.0f, t1 = 0.0f, t2 = 0.0f;
    #pragma unroll
    for (int w = 0; w < kWaves; ++w) {
      t0 += red[slot][w][0]; t1 += red[slot][w][1]; t2 += red[slot][w][2];
    }
    // w_j = nu_j / (mu * S_j) = (nu_j * n) / S'_j
    w0 = kNu0N * __builtin_amdgcn_rcpf(t0);
    w1 = kNu1N * __builtin_amdgcn_rcpf(t1);
    w2 = kNu2N * __builtin_amdgcn_rcpf(t2);
  }

  // ---- Output: n*(Gamma_2 - Gamma_0) = (k2*w2 - k0*w0) / denom(prev w) ----
  float o[kEPT];
  #pragma unroll
  for (int e = 0; e < kEPT; ++e) {
    float d = fmaf(k0[e], p0, fmaf(k1[e], p1, k2[e] * p2));
    float r = __builtin_amdgcn_rcpf(d);
    o[e] = (k2[e] * w2 - k0[e] * w0) * r;
  }
  float4* ov = (float4*)(out + (size_t)row * kN + tid * kEPT);
  #pragma unroll
  for (int c = 0; c < 4; ++c)
    ov[c] = make_float4(o[4 * c + 0], o[4 * c + 1], o[4 * c + 2], o[4 * c + 3]);
}

extern "C" void kernel_launch(void* const* d_in, const int* in_sizes, int n_in,
                              void* d_out, int out_size, void* d_ws, size_t ws_size,
                              hipStream_t stream) {
  (void)n_in; (void)d_ws; (void)ws_size; (void)out_size;
  const float* scores = (const float*)d_in[0];
  float* out = (float*)d_out;
  const int rows = in_sizes[0] / kN;   // 512
  soft_topk_sinkhorn<<<dim3(rows), dim3(kTPB), 0, stream>>>(scores, out);
}